// CombineUV_22952305230107
// MI455X (gfx1250) — compile-verified
//
#include <hip/hip_runtime.h>

typedef float v2f __attribute__((ext_vector_type(2)));
typedef float v8f __attribute__((ext_vector_type(8)));

#define DDIM 512
#define BDIM 512
#define SDIM 512
#define TILES_PER_BLOCK 8   // 8 waves per block, one 16-wide s-tile each

__global__ __launch_bounds__(256) void CombineUV_shortlist_wmma(
    const float* __restrict__ input,      // [B, D]
    const float* __restrict__ labels,     // [L, D]
    const float* __restrict__ weight,     // [L, D]
    const float* __restrict__ alpha,      // [1, D]
    const float* __restrict__ beta,       // [1, D]
    const float* __restrict__ bias,       // [L]
    const long long* __restrict__ shortlist, // [B, S] int64
    float* __restrict__ out)              // [B, S]
{
    __shared__ float sA[DDIM];  // input[b,:] * sigmoid(alpha)
    __shared__ float sB[DDIM];  // input[b,:] * sigmoid(beta)

    const int b   = blockIdx.x;
    const int tid = threadIdx.x;

    // Fold the sigmoid gates into the input row once per block (tiny vs. gather).
    for (int d = tid; d < DDIM; d += 256) {
        float x  = input[(size_t)b * DDIM + d];
        float sa = 1.0f / (1.0f + __expf(-alpha[d]));
        float sb = 1.0f / (1.0f + __expf(-beta[d]));
        sA[d] = x * sa;
        sB[d] = x * sb;
    }
    __syncthreads();

    const int wave = tid >> 5;
    const int lane = tid & 31;
    const int m    = lane & 15;   // M index (shortlist entry within tile)
    const int half = lane >> 4;   // K interleave half (ISA 16x4 fp32 A layout)

    const int s0 = (blockIdx.y * TILES_PER_BLOCK + wave) * 16;

    const long long idx = shortlist[(size_t)b * SDIM + s0 + m];
    const float* wrow = weight + (size_t)idx * DDIM;
    const float* lrow = labels + (size_t)idx * DDIM;

    v8f c = {0.f, 0.f, 0.f, 0.f, 0.f, 0.f, 0.f, 0.f};

    // A fragment: lane holds K = d+2*half (v0) and d+2*half+1 (v1) of row M=m.
    // B fragment: same K values of the gated input row, replicated across all N.
    const int koff = 2 * half;
    #pragma unroll 4
    for (int d = 0; d < DDIM; d += 4) {
        const int o = d + koff;
        v2f aw = *(const v2f*)(wrow + o);   // gathered refinement row (b64)
        v2f al = *(const v2f*)(lrow + o);   // gathered label-embedding row (b64)
        v2f bw = *(const v2f*)(&sA[o]);     // LDS b64
        v2f bl = *(const v2f*)(&sB[o]);     // LDS b64
        // D = A*B + C, fp32 exact; chain both dots into one accumulator.
        c = __builtin_amdgcn_wmma_f32_16x16x4_f32(false, aw, false, bw,
                                                  (short)0, c, false, false);
        c = __builtin_amdgcn_wmma_f32_16x16x4_f32(false, al, false, bl,
                                                  (short)0, c, false, false);
    }

    // All N columns identical: lane 0 holds M=0..7 in c[0..7], lane 16 holds M=8..15.
    if (lane == 0 || lane == 16) {
        const int mbase = half * 8;
        #pragma unroll
        for (int r = 0; r < 8; ++r) {
            const int s = s0 + mbase + r;
            const long long bi = shortlist[(size_t)b * SDIM + s];
            out[(size_t)b * SDIM + s] = c[r] + bias[bi];
        }
    }
}

extern "C" void kernel_launch(void* const* d_in, const int* in_sizes, int n_in,
                              void* d_out, int out_size, void* d_ws, size_t ws_size,
                              hipStream_t stream) {
    const float*     input     = (const float*)d_in[0];
    const float*     labels    = (const float*)d_in[1];
    const float*     weight    = (const float*)d_in[2];
    const float*     alpha     = (const float*)d_in[3];
    const float*     beta      = (const float*)d_in[4];
    const float*     bias      = (const float*)d_in[5];
    const long long* shortlist = (const long long*)d_in[6];
    float* out = (float*)d_out;

    dim3 grid(BDIM, SDIM / (16 * TILES_PER_BLOCK));  // (512, 4)
    CombineUV_shortlist_wmma<<<grid, 256, 0, stream>>>(
        input, labels, weight, alpha, beta, bias, shortlist, out);
}